// SpecAugment_24309514896021
// MI455X (gfx1250) — compile-verified
//
#include <hip/hip_runtime.h>
#include <hip/hip_bf16.h>

typedef unsigned int u32;
typedef unsigned long long u64;

// ---------------------------------------------------------------------------
// JAX Threefry-2x32 (exact): 20 rounds, ks2 = k0^k1^0x1BD11BDA
// ---------------------------------------------------------------------------
__device__ __forceinline__ u32 rotl32(u32 x, int r) { return (x << r) | (x >> (32 - r)); }

__device__ __forceinline__ void tf2x32(u32 k0, u32 k1, u32 c0, u32 c1, u32* o0, u32* o1) {
  const u32 ks2 = k0 ^ k1 ^ 0x1BD11BDAu;
  u32 x0 = c0 + k0, x1 = c1 + k1;
#define TF_R4(a, b, c, d)                                \
  x0 += x1; x1 = rotl32(x1, a); x1 ^= x0;                \
  x0 += x1; x1 = rotl32(x1, b); x1 ^= x0;                \
  x0 += x1; x1 = rotl32(x1, c); x1 ^= x0;                \
  x0 += x1; x1 = rotl32(x1, d); x1 ^= x0;
  TF_R4(13, 15, 26, 6);  x0 += k1;  x1 += ks2 + 1u;
  TF_R4(17, 29, 16, 24); x0 += ks2; x1 += k0 + 2u;
  TF_R4(13, 15, 26, 6);  x0 += k0;  x1 += k1 + 3u;
  TF_R4(17, 29, 16, 24); x0 += k1;  x1 += ks2 + 4u;
  TF_R4(13, 15, 26, 6);  x0 += ks2; x1 += k0 + 5u;
#undef TF_R4
  *o0 = x0; *o1 = x1;
}

// random_bits element i of a flat array of (even) size S under key (ka,kb).
__device__ __forceinline__ u32 rb(u32 ka, u32 kb, u32 i, u32 S) {
  const u32 half = S >> 1;
  const u32 c0 = (i < half) ? i : (i - half);
  u32 o0, o1;
  tf2x32(ka, kb, c0, c0 + half, &o0, &o1);
  return (i < half) ? o0 : o1;
}

// JAX randint core: offset = ((hi%span)*((2^16%span)^2%span) + lo%span) % span
__device__ __forceinline__ u32 randint_from_bits(u32 hi, u32 lo, u32 span) {
  u32 m = 65536u % span;
  m = (m * m) % span;
  return ((hi % span) * m + (lo % span)) % span;
}

// ---------------------------------------------------------------------------
// Kernel 1: per-sample mask parameters (exact JAX reproduction), 64 threads.
// ws layout (u32 units): [0..255]    fmask: 4 words/sample (3 used, 80 bits)
//                        [256..895]  tbeg[64][10] (int32)
//                        [896..1535] tend[64][10] (int32)
// ---------------------------------------------------------------------------
__global__ __launch_bounds__(64) void specaug_setup(const int* __restrict__ length,
                                                    u32* __restrict__ ws) {
  const int b = threadIdx.x;
  if (b >= 64) return;
  const u32 K0 = 0u, K1 = 42u;  // jax.random.key(42) -> data (0, 42)

  // split(key, 4): rows of reshaped threefry(key, iota(8))
  const u32 kfw0 = rb(K0, K1, 0, 8), kfw1 = rb(K0, K1, 1, 8);
  const u32 kfp0 = rb(K0, K1, 2, 8), kfp1 = rb(K0, K1, 3, 8);
  const u32 ktw0 = rb(K0, K1, 4, 8), ktw1 = rb(K0, K1, 5, 8);
  const u32 ktp0 = rb(K0, K1, 6, 8), ktp1 = rb(K0, K1, 7, 8);

  // randint internally splits its key into (k_hi, k_lo) via iota(4)
  const u32 fwh0 = rb(kfw0, kfw1, 0, 4), fwh1 = rb(kfw0, kfw1, 1, 4);
  const u32 fwl0 = rb(kfw0, kfw1, 2, 4), fwl1 = rb(kfw0, kfw1, 3, 4);
  const u32 fph0 = rb(kfp0, kfp1, 0, 4), fph1 = rb(kfp0, kfp1, 1, 4);
  const u32 fpl0 = rb(kfp0, kfp1, 2, 4), fpl1 = rb(kfp0, kfp1, 3, 4);
  const u32 twh0 = rb(ktw0, ktw1, 0, 4), twh1 = rb(ktw0, ktw1, 1, 4);
  const u32 twl0 = rb(ktw0, ktw1, 2, 4), twl1 = rb(ktw0, ktw1, 3, 4);

  int fw[2], fp[2];
#pragma unroll
  for (int j = 0; j < 2; ++j) {
    const u32 i = (u32)(b * 2 + j);  // shape (64,2) -> S=128
    fw[j] = (int)randint_from_bits(rb(fwh0, fwh1, i, 128), rb(fwl0, fwl1, i, 128), 28u);
    fp[j] = (int)randint_from_bits(rb(fph0, fph1, i, 128), rb(fpl0, fpl1, i, 128), 81u);
  }

  const float lenp1 = (float)length[b] + 1.0f;
  int tb[10], te[10];
#pragma unroll
  for (int j = 0; j < 10; ++j) {
    const u32 i = (u32)(b * 10 + j);  // shape (64,10) -> S=640
    const int tw = (int)randint_from_bits(rb(twh0, twh1, i, 640), rb(twl0, twl1, i, 640), 101u);
    const u32 bits = rb(ktp0, ktp1, i, 640);
    float u = __uint_as_float((bits >> 9) | 0x3F800000u) - 1.0f;  // jax.random.uniform
    u = fmaxf(u, 0.0f);
    const int tp = (int)floorf(u * lenp1);
    tb[j] = tp; te[j] = tp + tw;
  }

  u32 m[3] = {0u, 0u, 0u};
  for (int d = 0; d < 80; ++d) {
    const bool z = (d >= fp[0] && d < fp[0] + fw[0]) || (d >= fp[1] && d < fp[1] + fw[1]);
    if (z) m[d >> 5] |= (1u << (d & 31));
  }
  ws[b * 4 + 0] = m[0]; ws[b * 4 + 1] = m[1]; ws[b * 4 + 2] = m[2]; ws[b * 4 + 3] = 0u;
  int* tbw = (int*)(ws + 256);
  int* tew = (int*)(ws + 256 + 640);
#pragma unroll
  for (int j = 0; j < 10; ++j) { tbw[b * 10 + j] = tb[j]; tew[b * 10 + j] = te[j]; }
}

// ---------------------------------------------------------------------------
// Kernel 2: streaming mask-apply with depth-8 async global->LDS ring buffer.
// Grid (8, 64, 2): block = (1024-float t-chunk, batch b, 40-row d-half).
// 1024 blocks x 8 wave32 = 8192 waves; 7 x 512B in flight per wave ~= 28 MB
// aggregate, above the ~16 MB HBM bandwidth-delay product of 23.3 TB/s.
// Each wave consumes only LDS it loaded itself -> ASYNCcnt alone sequences.
// ---------------------------------------------------------------------------
#define NBUF 8   // ring depth (power of 2); 8 x 4KB = 32KB LDS
#define NROW 40  // rows per block

__device__ __forceinline__ void async_ld_b128(u32 lds_off, u32 voff, u64 sbase) {
  // GVS mode: reads 16 bytes at the 64-bit base in sbase plus the per-lane offset in voff into LDS at lds_off
  asm volatile("global_load_async_to_lds_b128 %0, %1, %2"
               :: "v"(lds_off), "v"(voff), "s"(sbase) : "memory");
}

__global__ __launch_bounds__(256) void specaug_apply(const float* __restrict__ in,
                                                     float* __restrict__ out,
                                                     const u32* __restrict__ ws) {
  __shared__ __align__(16) float tile[NBUF][1024];
  const int b   = blockIdx.y;
  const int cx  = blockIdx.x;             // 8 chunks of 1024 along T
  const int d0  = blockIdx.z * NROW;      // 2 halves of 40 rows along D
  const int tid = threadIdx.x;
  const int t0  = cx * 1024 + tid * 4;

  // per-sample masks
  const u32 fm0 = ws[b * 4 + 0], fm1 = ws[b * 4 + 1], fm2 = ws[b * 4 + 2];
  const int* tbw = (const int*)(ws + 256) + b * 10;
  const int* tew = (const int*)(ws + 256 + 640) + b * 10;

  u32 km[4];
#pragma unroll
  for (int e = 0; e < 4; ++e) {
    const int t = t0 + e;
    bool z = false;
#pragma unroll
    for (int j = 0; j < 10; ++j) z = z || (t >= tbw[j] && t < tew[j]);
    km[e] = z ? 0u : 0xFFFFFFFFu;
  }

  const size_t chunkOff = (size_t)b * 80u * 8192u + (size_t)cx * 1024u;  // floats
  const u32 voff = (u32)(tid * 16);
  // low 32 bits of the generic address of a __shared__ object == LDS byte offset
  const u32 ldsBase = (u32)(uintptr_t)&tile[0][tid * 4];

  // prologue: stage rows d0 .. d0+6
#pragma unroll
  for (int p = 0; p < NBUF - 1; ++p) {
    async_ld_b128(ldsBase + (u32)(p << 12), voff,
                  (u64)(uintptr_t)(in + chunkOff + (size_t)(d0 + p) * 8192u));
  }

  for (int k = 0; k < NROW; ++k) {
    const int d = d0 + k;
    if (k < NROW - (NBUF - 1)) {
      // keep 8 loads outstanding; oldest (row d) retires at cnt<=7 (in-order)
      async_ld_b128(ldsBase + (u32)(((k + NBUF - 1) & (NBUF - 1)) << 12), voff,
                    (u64)(uintptr_t)(in + chunkOff + (size_t)(d + NBUF - 1) * 8192u));
      asm volatile("s_wait_asynccnt 0x7" ::: "memory");
    } else if (k == NROW - (NBUF - 1)) {
      asm volatile("s_wait_asynccnt 0x0" ::: "memory");  // drain tail
    }
    const uint4 vb = *(const uint4*)&tile[k & (NBUF - 1)][tid * 4];  // ds_load_b128
    const u32 fmw   = (d < 32) ? fm0 : ((d < 64) ? fm1 : fm2);
    const u32 keepF = ((fmw >> (d & 31)) & 1u) ? 0u : 0xFFFFFFFFu;
    uint4 r;
    r.x = vb.x & km[0] & keepF;
    r.y = vb.y & km[1] & keepF;
    r.z = vb.z & km[2] & keepF;
    r.w = vb.w & km[3] & keepF;
    *(uint4*)(out + chunkOff + (size_t)d * 8192u + (size_t)(tid * 4)) = r;  // global_store_b128
  }
}

// ---------------------------------------------------------------------------
extern "C" void kernel_launch(void* const* d_in, const int* in_sizes, int n_in,
                              void* d_out, int out_size, void* d_ws, size_t ws_size,
                              hipStream_t stream) {
  (void)in_sizes; (void)n_in; (void)out_size; (void)ws_size;
  const float* in     = (const float*)d_in[0];
  const int*   length = (const int*)d_in[1];
  float*       out    = (float*)d_out;
  u32*         ws     = (u32*)d_ws;

  specaug_setup<<<1, 64, 0, stream>>>(length, ws);
  specaug_apply<<<dim3(8, 64, 2), 256, 0, stream>>>(in, out, ws);
}